// TSALoss_56066503082324
// MI455X (gfx1250) — compile-verified
//
#include <hip/hip_runtime.h>
#include <stdint.h>

typedef __attribute__((ext_vector_type(2))) float v2f;
typedef __attribute__((ext_vector_type(8))) float v8f;

#define B_N 16384
#define D_N 16
#define K_NBR 25
#define LIST_LEN 26          // K+1 (self included, dropped later)
#define PIT 32               // power-iteration steps
#define LAMBDA_TSA 0.1f
#define COV_DENOM (1.0f / (24.0f + 1e-8f))
#define KNN_WAVES 4          // anchor-tile waves sharing one LDS-staged candidate tile

// ------------------------------------------------------------------
// CDNA5 async global->LDS staging (ASYNCcnt path), with portable fallback
// Builtin signature (from hipcc diagnostic): param0 = v4i addrspace(1)*,
// param1 = v4i addrspace(3)*, then imm offset, imm cpol.
// ------------------------------------------------------------------
#if defined(__HIP_DEVICE_COMPILE__) && __has_builtin(__builtin_amdgcn_global_load_async_to_lds_b128)
#define HAVE_ASYNC 1
typedef int v4i __attribute__((vector_size(16)));
typedef __attribute__((address_space(1))) v4i gv4i;
typedef __attribute__((address_space(3))) v4i lv4i;
template <int OFF>
__device__ __forceinline__ void async_b128(const float* g, float* l) {
    // the one imm OFF is added to both the LDS and the global address
    __builtin_amdgcn_global_load_async_to_lds_b128(
        (gv4i*)(unsigned long long)(uintptr_t)g,
        (lv4i*)(unsigned)(uintptr_t)l,   // low 32 bits of generic addr = LDS offset
        OFF, 0);
}
#if __has_builtin(__builtin_amdgcn_s_wait_asynccnt)
#define WAIT_ASYNC(n) __builtin_amdgcn_s_wait_asynccnt(n)
#else
#define WAIT_ASYNC(n) asm volatile("s_wait_asynccnt %0" ::"n"(n) : "memory")
#endif
#else
#define HAVE_ASYNC 0
template <int OFF>
__device__ __forceinline__ void async_b128(const float* g, float* l) {
    *(float4*)((char*)l + OFF) = *(const float4*)((const char*)g + OFF);
}
#define WAIT_ASYNC(n) ((void)0)
#endif

// stage one 16x16 candidate tile (1 KiB) + its 16 sq values into LDS
__device__ __forceinline__ void prefetch_tile(const float* __restrict__ raw,
                                              const float* __restrict__ sq,
                                              float* cdst, float* sdst,
                                              int jt, int lane) {
    const float* g = raw + jt * 256 + lane * 4;   // 16 rows x 64B, lane covers 16B units
    async_b128<0>(g, cdst + lane * 4);
    async_b128<512>(g, cdst + lane * 4);
    if (lane < 4) async_b128<0>(sq + jt * 16 + lane * 4, sdst + lane * 4);
}

// ---------------------------------------------------------------- utilities
__device__ __forceinline__ float half_reduce(float v) {
    v += __shfl_xor(v, 1, 32);
    v += __shfl_xor(v, 2, 32);
    v += __shfl_xor(v, 4, 32);
    v += __shfl_xor(v, 8, 32);
    return v;
}

// ---------------------------------------------------------------- kernel 0
__global__ void zero_kernel(float* __restrict__ out) {
    if (threadIdx.x == 0 && blockIdx.x == 0) out[0] = 0.0f;
}

// ---------------------------------------------------------------- kernel 1: recon MSE
__global__ void __launch_bounds__(256) recon_kernel(const float* __restrict__ outputs,
                                                    const float* __restrict__ targets,
                                                    float* __restrict__ out) {
    int i = blockIdx.x * blockDim.x + threadIdx.x;
    float d = outputs[i] - targets[i];
    float v = d * d;
    v = half_reduce(v);
    v += __shfl_xor(v, 16, 32);
    __shared__ float part[8];
    int w = threadIdx.x >> 5;
    if ((threadIdx.x & 31) == 0) part[w] = v;
    __syncthreads();
    if (threadIdx.x == 0) {
        float s = 0.0f;
        for (int j = 0; j < 8; ++j) s += part[j];
        atomicAdd(out, s * (1.0f / (float)(B_N * D_N)));
    }
}

// ---------------------------------------------------------------- kernel 2: row squared norms
__global__ void __launch_bounds__(256) sq_kernel(const float* __restrict__ raw,
                                                 float* __restrict__ sq) {
    int i = blockIdx.x * blockDim.x + threadIdx.x;
    const float4* r = (const float4*)(raw + i * D_N);
    float s = 0.0f;
    #pragma unroll
    for (int t = 0; t < 4; ++t) {
        float4 v = r[t];
        s += v.x * v.x + v.y * v.y + v.z * v.z + v.w * v.w;
    }
    sq[i] = s;
}

// ---------------------------------------------------------------- kernel 3: WMMA kNN (top-26)
// 4 waves per block; each wave owns one 16-anchor tile. Wave 0 double-buffers
// candidate tiles into LDS with async B128 loads (ASYNCcnt-fenced); all waves
// chain 4x v_wmma_f32_16x16x4_f32 per tile (M=candidates, N=anchors) and keep
// per-lane register-resident sorted top-26 lists.
__global__ void __launch_bounds__(32 * KNN_WAVES) knn_kernel(const float* __restrict__ raw,
                                                             const float* __restrict__ sq,
                                                             int* __restrict__ nbrs) {
    const int tid  = threadIdx.x;
    const int wave = tid >> 5;
    const int lane = tid & 31;
    const int hf   = lane >> 4;
    const int n    = lane & 15;
    const int anchor = (blockIdx.x * KNN_WAVES + wave) * 16 + n;

    __shared__ __align__(16) float cbuf[2][256];   // double-buffered candidate tile
    __shared__ __align__(16) float sqb[2][16];     // candidate squared norms
    __shared__ float ld[32 * KNN_WAVES * LIST_LEN];
    __shared__ int   li[32 * KNN_WAVES * LIST_LEN];

    // fixed B fragments (anchor matrix): lanes<16 hold k={4t,4t+1}, lanes>=16 k={4t+2,4t+3}
    v2f bfrag[4];
    #pragma unroll
    for (int t = 0; t < 4; ++t)
        bfrag[t] = *(const v2f*)(raw + anchor * D_N + 4 * t + 2 * hf);
    const float sqi = sq[anchor];

    float best_d[LIST_LEN];
    int   best_i[LIST_LEN];
    #pragma unroll
    for (int s = 0; s < LIST_LEN; ++s) { best_d[s] = 3.0e38f; best_i[s] = 0; }

    if (wave == 0) prefetch_tile(raw, sq, cbuf[0], sqb[0], 0, lane);

    const int NT = B_N / 16;
    for (int jt = 0; jt < NT; ++jt) {
        const int p = jt & 1;
        if (wave == 0) {
            if (jt + 1 < NT) {
                prefetch_tile(raw, sq, cbuf[p ^ 1], sqb[p ^ 1], jt + 1, lane);
                WAIT_ASYNC(3);   // in-order completion => the 3 ops for buffer p landed
            } else {
                WAIT_ASYNC(0);
            }
        }
        __syncthreads();                         // buffer p published to all waves

        v8f acc = {};
        #pragma unroll
        for (int t = 0; t < 4; ++t) {
            v2f a = *(const v2f*)&cbuf[p][n * 16 + 4 * t + 2 * hf];   // ds_load_b64
            acc = __builtin_amdgcn_wmma_f32_16x16x4_f32(
                      false, a, false, bfrag[t], (short)0, acc, false, false);
        }
        float4 s0 = *(const float4*)&sqb[p][8 * hf];
        float4 s1 = *(const float4*)&sqb[p][8 * hf + 4];
        float sqj[8] = {s0.x, s0.y, s0.z, s0.w, s1.x, s1.y, s1.z, s1.w};

        #pragma unroll
        for (int r = 0; r < 8; ++r) {
            float d  = sqi + sqj[r] - 2.0f * acc[r];
            int  idx = jt * 16 + 8 * hf + r;
            if (d < best_d[LIST_LEN - 1]) {      // rare after warm-up
                best_d[LIST_LEN - 1] = d;
                best_i[LIST_LEN - 1] = idx;
                #pragma unroll
                for (int s = LIST_LEN - 1; s > 0; --s) {
                    if (best_d[s] < best_d[s - 1]) {
                        float td = best_d[s]; best_d[s] = best_d[s - 1]; best_d[s - 1] = td;
                        int   ti = best_i[s]; best_i[s] = best_i[s - 1]; best_i[s - 1] = ti;
                    }
                }
            }
        }
        __syncthreads();                         // reads of buffer p done before reuse
    }

    // merge the two half-wave lists of each anchor; drop element 0 (= self)
    #pragma unroll
    for (int s = 0; s < LIST_LEN; ++s) {
        ld[tid * LIST_LEN + s] = best_d[s];
        li[tid * LIST_LEN + s] = best_i[s];
    }
    __syncthreads();
    if (lane < 16) {
        const float* da = &ld[tid * LIST_LEN];
        const int*   ia = &li[tid * LIST_LEN];
        const float* db = &ld[(tid + 16) * LIST_LEN];
        const int*   ib = &li[(tid + 16) * LIST_LEN];
        int pa = 0, pb = 0;
        for (int s = 0; s < LIST_LEN; ++s) {
            int vi;
            if (da[pa] <= db[pb]) { vi = ia[pa]; ++pa; }
            else                  { vi = ib[pb]; ++pb; }
            if (s > 0) nbrs[anchor * K_NBR + (s - 1)] = vi;
        }
    }
}

// ---------------------------------------------------------------- kernel 4: covariance + power iteration
// One wave per anchor. Cz = Zc^T Zc via WMMA f32 16x16x4 (K padded 25->28);
// A and B fragments are the same per-lane LDS reads, so one v2f feeds both
// operands. Top eigenvector by power iteration; term = 2 - 2*(u.v)^2.
__global__ void __launch_bounds__(32) tsa_kernel(const float* __restrict__ raw,
                                                 const float* __restrict__ latent,
                                                 const int* __restrict__ nbrs,
                                                 float* __restrict__ out) {
    const int lane = threadIdx.x;
    const int hf   = lane >> 4;
    const int n    = lane & 15;
    const int b    = blockIdx.x;

    __shared__ int   nb[K_NBR];
    __shared__ float zc[28 * 16];
    __shared__ float xc[28 * 16];
    __shared__ float mz[16];
    __shared__ float mx[16];
    __shared__ float uz[16];

    if (lane < K_NBR) nb[lane] = nbrs[b * K_NBR + lane];
    __syncthreads();

    for (int e = lane; e < 28 * 16; e += 32) {
        int k = e >> 4, i = e & 15;
        float zv = 0.0f, xv = 0.0f;
        if (k < K_NBR) {
            int j = nb[k];
            zv = latent[j * D_N + i];
            xv = raw[j * D_N + i];
        }
        zc[e] = zv;
        xc[e] = xv;
    }
    __syncthreads();
    if (lane < 16) {
        float sz = 0.0f, sx = 0.0f;
        for (int k = 0; k < K_NBR; ++k) { sz += zc[k * 16 + lane]; sx += xc[k * 16 + lane]; }
        mz[lane] = sz * (1.0f / (float)K_NBR);
        mx[lane] = sx * (1.0f / (float)K_NBR);
    }
    __syncthreads();
    for (int e = lane; e < K_NBR * 16; e += 32) {
        int i = e & 15;
        zc[e] -= mz[i];
        xc[e] -= mx[i];
    }
    __syncthreads();

    v8f cz = {}, cx = {};
    #pragma unroll
    for (int t = 0; t < 7; ++t) {
        int k0 = 4 * t + 2 * hf;
        v2f fz, fx;
        fz.x = zc[(k0    ) * 16 + n];
        fz.y = zc[(k0 + 1) * 16 + n];
        fx.x = xc[(k0    ) * 16 + n];
        fx.y = xc[(k0 + 1) * 16 + n];
        cz = __builtin_amdgcn_wmma_f32_16x16x4_f32(false, fz, false, fz, (short)0, cz, false, false);
        cx = __builtin_amdgcn_wmma_f32_16x16x4_f32(false, fx, false, fx, (short)0, cx, false, false);
    }
    #pragma unroll
    for (int r = 0; r < 8; ++r) { cz[r] *= COV_DENOM; cx[r] *= COV_DENOM; }

    float dot = 0.0f;
    for (int pass = 0; pass < 2; ++pass) {
        v8f c = (pass == 0) ? cz : cx;
        float x = 1.0f + 0.03125f * (float)n;
        for (int it = 0; it < PIT; ++it) {
            float y[8];
            #pragma unroll
            for (int r = 0; r < 8; ++r)
                y[r] = half_reduce(c[r] * x);     // y[m] = sum_n C[m][n] x[n]
            __syncthreads();
            if (n == 0) {
                #pragma unroll
                for (int r = 0; r < 8; ++r) mz[8 * hf + r] = y[r];
            }
            __syncthreads();
            x = mz[n];
            float s = half_reduce(x * x);
            x *= rsqrtf(s + 1e-30f);
        }
        if (pass == 0) {
            __syncthreads();
            if (lane < 16) uz[lane] = x;
            __syncthreads();
        } else {
            dot = half_reduce(x * uz[n]);
        }
    }
    if (lane == 0) {
        float term = 2.0f - 2.0f * dot * dot;
        atomicAdd(out, term * (LAMBDA_TSA / (float)B_N));
    }
}

// ---------------------------------------------------------------- launcher
extern "C" void kernel_launch(void* const* d_in, const int* in_sizes, int n_in,
                              void* d_out, int out_size, void* d_ws, size_t ws_size,
                              hipStream_t stream) {
    const float* outputs = (const float*)d_in[0];
    const float* targets = (const float*)d_in[1];
    const float* latent  = (const float*)d_in[2];
    const float* raw     = (const float*)d_in[3];
    float* out = (float*)d_out;

    float* sq   = (float*)d_ws;                                   // 16384 floats
    int*   nbrs = (int*)((char*)d_ws + B_N * sizeof(float));      // 16384*25 ints

    zero_kernel <<<1, 1, 0, stream>>>(out);
    recon_kernel<<<(B_N * D_N) / 256, 256, 0, stream>>>(outputs, targets, out);
    sq_kernel   <<<B_N / 256, 256, 0, stream>>>(raw, sq);
    knn_kernel  <<<B_N / (16 * KNN_WAVES), 32 * KNN_WAVES, 0, stream>>>(raw, sq, nbrs);
    tsa_kernel  <<<B_N, 32, 0, stream>>>(raw, latent, nbrs, out);
}